// RelationGraphAttentionLayer_36618891165914
// MI455X (gfx1250) — compile-verified
//
#include <hip/hip_runtime.h>
#include <math.h>

typedef __attribute__((ext_vector_type(2))) float v2f;
typedef __attribute__((ext_vector_type(8))) float v8f;

#define N_NODES 500
#define DIN 100
#define DE  200
#define DOUT 100
#define KTOT 300           // DE + DIN (ent-proj K plus folded-in e_dst K)
#define KSTEPS 75          // KTOT / 4 (f32 WMMA is 16x16x4)
#define CH 32              // j-chunk per iteration
#define ASTR 308           // A_lds row stride (>=KTOT, mult of 4, bank-friendly)
#define ESTR 116           // e_lds row stride (112 cols + pad, conflict-free)
#define NT 224             // 7 waves: one 16-wide o-tile per wave (7*16=112>=100)
#define NEG_BIG (-9000000000000000.0f)
#define SLOPE 0.2f

__global__ __launch_bounds__(NT) void gat_fused_kernel(
    const float* __restrict__ rel,   // [500,100]
    const float* __restrict__ emb,   // [500,500,200]
    const int*   __restrict__ adj,   // [500,500]
    const float* __restrict__ a,     // [100,400]
    const float* __restrict__ a2,    // [100]
    float* __restrict__ out)         // [500,100]
{
    __shared__ __align__(16) float A_lds[CH * ASTR];   // 39424 B
    __shared__ __align__(16) float e_lds[CH * ESTR];   // 14848 B
    __shared__ float es_lds[112];
    __shared__ float a2_lds[112];
    __shared__ float acc_lds[112];
    __shared__ float sc_lds[CH];
    __shared__ float w_lds[CH];
    __shared__ float m_run, s_run, scale_sh;

    const int t    = threadIdx.x;
    const int i    = blockIdx.x;
    const int lane = t & 31;
    const int wv   = t >> 5;        // 0..6 : o-tile owner
    const int row  = lane & 15;
    const int half = lane >> 4;

    // ---------- per-block setup ----------
    if (t < 112) {
        a2_lds[t]  = (t < DOUT) ? a2[t] : 0.0f;
        acc_lds[t] = 0.0f;
        float s = 0.0f;
        if (t < DOUT) {                      // e_src[i, o] = rel[i]·a_src[o]
            const float* ar = a + (size_t)t * 400;
            const float* rr = rel + (size_t)i * DIN;
            #pragma unroll 4
            for (int d = 0; d < DIN; ++d) s += rr[d] * ar[d];
        }
        es_lds[t] = s;
    }
    if (t == 0) { m_run = NEG_BIG; s_run = 0.0f; }

    // ---------- B fragments in registers (K=300 x this wave's 16 o's) ----------
    // layout: frag[s].{x,y} = B[4s + 2*half + {0,1}][o_base + row]
    v2f bfrag[KSTEPS];
    {
        const int o = wv * 16 + row;
        #pragma unroll
        for (int s = 0; s < KSTEPS; ++s) {
            const int k0 = 4 * s + 2 * half;
            float b0 = 0.0f, b1 = 0.0f;
            if (o < DOUT) {
                int k = k0;      // B_ext[k][o]: k<200 -> a_ent[o][k]=a[o][200+k]
                b0 = a[(size_t)o * 400 + (k < DE ? DE + k : k - DIN)];
                k = k0 + 1;      //            k>=200 -> a_dst[o][k-200]=a[o][k-100]
                b1 = a[(size_t)o * 400 + (k < DE ? DE + k : k - DIN)];
            }
            bfrag[s].x = b0; bfrag[s].y = b1;
        }
    }
    __syncthreads();

    // ---------- stream over j in chunks of 32 (flash-style online softmax) ----------
    for (int j0 = 0; j0 < N_NODES; j0 += CH) {
        // stage A_ext chunk: cols [0,200)=emb[i][j], [200,300)=rel[j]
        for (int u = t; u < CH * (DE / 4); u += NT) {
            const int jj = u / (DE / 4), k4 = u % (DE / 4), jg = j0 + jj;
            float4 v = make_float4(0.f, 0.f, 0.f, 0.f);
            if (jg < N_NODES)
                v = *(const float4*)(emb + ((size_t)i * N_NODES + jg) * DE + k4 * 4);
            *(float4*)&A_lds[jj * ASTR + k4 * 4] = v;
        }
        for (int u = t; u < CH * (DIN / 4); u += NT) {
            const int jj = u / (DIN / 4), d4 = u % (DIN / 4), jg = j0 + jj;
            float4 v = make_float4(0.f, 0.f, 0.f, 0.f);
            if (jg < N_NODES)
                v = *(const float4*)(rel + (size_t)jg * DIN + d4 * 4);
            *(float4*)&A_lds[jj * ASTR + DE + d4 * 4] = v;
        }
        __syncthreads();

        // WMMA GEMM: e_tile = A_ext x B_ext + e_src  (fp32 16x16x4)
        {
            const int ob = wv * 16;
            const float esv = es_lds[ob + row];
            #pragma unroll
            for (int jt = 0; jt < 2; ++jt) {
                v8f c;
                #pragma unroll
                for (int v = 0; v < 8; ++v) c[v] = esv;
                const float* arow = &A_lds[(jt * 16 + row) * ASTR + 2 * half];
                #pragma unroll
                for (int s = 0; s < KSTEPS; ++s) {
                    v2f av = *(const v2f*)(arow + 4 * s);
                    c = __builtin_amdgcn_wmma_f32_16x16x4_f32(
                        false, av, false, bfrag[s], (short)0, c, false, false);
                }
                #pragma unroll
                for (int v = 0; v < 8; ++v) {
                    const int m = jt * 16 + v + 8 * half;   // C layout: lanes16-31 -> M+8
                    e_lds[m * ESTR + ob + row] = c[v];
                }
            }
        }
        __syncthreads();

        // scores: e2[j] = leaky(e_tile[j]·a2), mask with adj; pads -> -inf
        if (t < 128) {
            const int j = t >> 2, q = t & 3;
            float p = 0.f;
            const float* er = &e_lds[j * ESTR];
            #pragma unroll 5
            for (int o = q * 25; o < q * 25 + 25; ++o) p += er[o] * a2_lds[o];
            p += __shfl_xor(p, 1, 32);
            p += __shfl_xor(p, 2, 32);
            if (q == 0) {
                const int jg = j0 + j;
                float sc;
                if (jg < N_NODES) {
                    const float e2 = (p > 0.f) ? p : SLOPE * p;
                    sc = (adj[(size_t)i * N_NODES + jg] > 0) ? e2 : NEG_BIG;
                } else {
                    sc = -__builtin_inff();
                }
                sc_lds[j] = sc;
            }
        }
        __syncthreads();

        // online softmax update (wave 0)
        if (t < 32) {
            const float sc = sc_lds[t];
            float mt = sc;
            for (int off = 16; off; off >>= 1) mt = fmaxf(mt, __shfl_xor(mt, off, 32));
            const float mo = m_run;
            const float mn = fmaxf(mo, mt);
            const float w  = __expf(sc - mn);      // -inf pads -> 0
            float sw = w;
            for (int off = 16; off; off >>= 1) sw += __shfl_xor(sw, off, 32);
            w_lds[t] = w;
            if (t == 0) {
                const float scl = __expf(mo - mn);
                s_run    = s_run * scl + sw;
                m_run    = mn;
                scale_sh = scl;
            }
        }
        __syncthreads();

        // acc[o] = acc[o]*scale + sum_j w[j]*e_tile[j][o]
        if (t < 112) {
            float av = acc_lds[t] * scale_sh;
            #pragma unroll 8
            for (int j = 0; j < CH; ++j) av += w_lds[j] * e_lds[j * ESTR + t];
            acc_lds[t] = av;
        }
        __syncthreads();
    }

    // ---------- finalize: h = acc/s ; out = elu(h) ----------
    if (t < DOUT) {
        const float h = acc_lds[t] / s_run;
        out[(size_t)i * DOUT + t] = (h > 0.f) ? h : (__expf(h) - 1.0f);
    }
}

extern "C" void kernel_launch(void* const* d_in, const int* in_sizes, int n_in,
                              void* d_out, int out_size, void* d_ws, size_t ws_size,
                              hipStream_t stream) {
    (void)in_sizes; (void)n_in; (void)out_size; (void)d_ws; (void)ws_size;
    const float* rel = (const float*)d_in[0];
    const float* emb = (const float*)d_in[1];
    const int*   adj = (const int*)d_in[2];
    const float* a   = (const float*)d_in[3];
    const float* a2  = (const float*)d_in[4];
    float* out = (float*)d_out;
    gat_fused_kernel<<<dim3(N_NODES), dim3(NT), 0, stream>>>(rel, emb, adj, a, a2, out);
}